// SmallestAdditionTransformer_37752762531954
// MI455X (gfx1250) — compile-verified
//
#include <hip/hip_runtime.h>
#include <math.h>

#define N_LAYERS 4
#define T        34
#define TP       48      // T padded to 3 WMMA tiles of 16
#define PSTRIDE  50      // score row stride (floats): conflict-free column scans, 8B aligned
#define VOCAB    14
#define DM       6
#define HD       3

typedef float v2f __attribute__((ext_vector_type(2)));
typedef float v4f __attribute__((ext_vector_type(4)));
typedef float v8f __attribute__((ext_vector_type(8)));

// One workgroup (64 threads = 2 waves) per sequence.
// Wave 0 = token head, wave 1 = position head.
__global__ __launch_bounds__(64)
void addtrans_kernel(const int* __restrict__ idx,
                     const float* __restrict__ tok_emb,
                     const float* __restrict__ pos_enc,
                     const float* __restrict__ ln_w, const float* __restrict__ ln_b,
                     const float* __restrict__ q1w, const float* __restrict__ k1w,
                     const float* __restrict__ v1w,
                     const float* __restrict__ q2w, const float* __restrict__ k2w,
                     const float* __restrict__ v2w,
                     const float* __restrict__ out_w,
                     const float* __restrict__ lnf_w, const float* __restrict__ lnf_b,
                     const float* __restrict__ head_w,
                     float* __restrict__ out)
{
    __shared__ float xs[TP][DM];
    __shared__ float hs[TP][DM];
    __shared__ float qs[2][TP][4];            // [3] == 0 (K=3 zero pad)
    __shared__ float ks_[2][TP][4];           // [3] == 0
    __shared__ float vs[2][TP][4];            // [3] == 0 (doubles as N-pad source)
    __shared__ float os[2][TP][4];
    __shared__ float Ps[2][TP][PSTRIDE];      // scores -> probabilities, per wave

    const int b    = blockIdx.x;
    const int tid  = threadIdx.x;
    const int w    = tid >> 5;                // wave id = head id
    const int lane = tid & 31;
    const int hi16 = lane >> 4;               // K-half selector for WMMA fragments
    const int l16  = lane & 15;
    const int kofs = hi16 * 2;                // fragment K offset (0 or 2)
    const int colc = (l16 < 3) ? l16 : 3;     // clamped N index; [3] is the stored zero

    __builtin_prefetch(idx + (size_t)b * T, 0, 1);   // global_prefetch_b8

    // ---- embedding: x = [tok_emb[idx], pos_enc] ----
    for (int t = tid; t < T; t += 64) {
        int tok = idx[(size_t)b * T + t];
#pragma unroll
        for (int c = 0; c < 3; ++c) {
            xs[t][c]     = tok_emb[tok * 3 + c];
            xs[t][3 + c] = pos_enc[t * 3 + c];
        }
    }
    __syncthreads();

    const float rs3 = 0.57735026918962576f;   // 1/sqrt(HEAD_DIM)

    for (int l = 0; l < N_LAYERS; ++l) {
        // ---- layernorm over D=6 (one thread per position) ----
        if (tid < TP) {
            if (tid < T) {
                float m = 0.f;
#pragma unroll
                for (int c = 0; c < DM; ++c) m += xs[tid][c];
                m *= (1.0f / DM);
                float var = 0.f;
#pragma unroll
                for (int c = 0; c < DM; ++c) { float d = xs[tid][c] - m; var += d * d; }
                var *= (1.0f / DM);
                float r = rsqrtf(var + 1e-5f);
#pragma unroll
                for (int c = 0; c < DM; ++c)
                    hs[tid][c] = (xs[tid][c] - m) * r * ln_w[l * DM + c] + ln_b[l * DM + c];
            } else {
#pragma unroll
                for (int c = 0; c < DM; ++c) hs[tid][c] = 0.f;
            }
        }
        __syncthreads();

        // ---- per-head q/k/v: 3x3 matvecs; wave w owns head w; one b128 store per row ----
        const float* qw = (w == 0 ? q1w : q2w) + l * 9;
        const float* kw = (w == 0 ? k1w : k2w) + l * 9;
        const float* vw = (w == 0 ? v1w : v2w) + l * 9;
        const int hoff = w * HD;
        for (int t = lane; t < TP; t += 32) {
            float h0 = hs[t][hoff], h1 = hs[t][hoff + 1], h2 = hs[t][hoff + 2];
            float valid = (t < T) ? 1.0f : 0.0f;
            v4f qv, kv, vv;
#pragma unroll
            for (int i = 0; i < HD; ++i) {
                qv[i] = (qw[i*3]*h0 + qw[i*3+1]*h1 + qw[i*3+2]*h2) * valid;
                kv[i] = (kw[i*3]*h0 + kw[i*3+1]*h1 + kw[i*3+2]*h2) * valid;
                vv[i] = (vw[i*3]*h0 + vw[i*3+1]*h1 + vw[i*3+2]*h2) * valid;
            }
            qv[3] = 0.f; kv[3] = 0.f; vv[3] = 0.f;
            *(v4f*)&qs[w][t][0]  = qv;
            *(v4f*)&ks_[w][t][0] = kv;
            *(v4f*)&vs[w][t][0]  = vv;
        }
        // same-wave LDS ops are in-order: no barrier needed for per-wave phases

        // ---- S = Q K^T via V_WMMA_F32_16X16X4_F32 (lower-triangular tiles only) ----
        for (int mi = 0; mi < 3; ++mi) {
            const int am = mi * 16 + l16;
            v2f a = *(const v2f*)&qs[w][am][kofs];        // one ds_load_b64, branchless
            for (int ni = 0; ni <= mi; ++ni) {
                const int bn = ni * 16 + l16;
                v2f bf = *(const v2f*)&ks_[w][bn][kofs];  // one ds_load_b64, branchless
                v8f c = {0.f, 0.f, 0.f, 0.f, 0.f, 0.f, 0.f, 0.f};
                c = __builtin_amdgcn_wmma_f32_16x16x4_f32(false, a, false, bf,
                                                          (short)0, c, false, false);
#pragma unroll
                for (int r = 0; r < 8; ++r)
                    Ps[w][mi * 16 + hi16 * 8 + r][ni * 16 + l16] = c[r];
            }
        }

        // ---- causal softmax per row (scale 1/sqrt(3)); branchless, zero-pads P ----
        for (int m = lane; m < TP; m += 32) {
            const bool rowok = (m < T);
            float mx = -3.4e38f;
            for (int n = 0; n < TP; ++n) {
                float s = Ps[w][m][n] * rs3;
                bool valid = rowok && (n <= m);
                mx = valid ? fmaxf(mx, s) : mx;
            }
            float sum = 0.f;
            for (int n = 0; n < TP; ++n) {
                float s = Ps[w][m][n] * rs3;
                bool valid = rowok && (n <= m);
                float e = valid ? __expf(s - mx) : 0.f;
                Ps[w][m][n] = e;
                sum += e;
            }
            float inv = (sum > 0.f) ? (1.0f / sum) : 0.f;
            for (int n = 0; n < TP; ++n) Ps[w][m][n] *= inv;
        }

        // ---- O = P V via WMMA, accumulating K-chunks of 4 ----
        for (int mi = 0; mi < 3; ++mi) {
            v8f acc = {0.f, 0.f, 0.f, 0.f, 0.f, 0.f, 0.f, 0.f};
            const int am = mi * 16 + l16;
            int kmax = 4 * (mi + 1);                  // causal: later chunks are all-zero P
            if (kmax > (T + 3) / 4) kmax = (T + 3) / 4;
            for (int kc = 0; kc < kmax; ++kc) {
                const int nb = kc * 4;
                v2f a = *(const v2f*)&Ps[w][am][nb + kofs];   // ds_load_b64, branchless
                v2f bf;
                bf.x = vs[w][nb + kofs + 0][colc];            // colc==3 -> stored zero
                bf.y = vs[w][nb + kofs + 1][colc];
                acc = __builtin_amdgcn_wmma_f32_16x16x4_f32(false, a, false, bf,
                                                            (short)0, acc, false, false);
            }
            if (l16 < HD) {
#pragma unroll
                for (int r = 0; r < 8; ++r)
                    os[w][mi * 16 + hi16 * 8 + r][l16] = acc[r];
            }
        }
        __syncthreads();

        // ---- x += out_w @ concat(o1, o2) ----
        if (tid < T) {
            float u[DM];
#pragma unroll
            for (int i = 0; i < HD; ++i) { u[i] = os[0][tid][i]; u[HD + i] = os[1][tid][i]; }
#pragma unroll
            for (int i = 0; i < DM; ++i) {
                float acc = 0.f;
#pragma unroll
                for (int j = 0; j < DM; ++j) acc += out_w[l * 36 + i * 6 + j] * u[j];
                xs[tid][i] += acc;
            }
        }
        __syncthreads();
    }

    // ---- final layernorm + vocab head ----
    if (tid < T) {
        float m = 0.f;
#pragma unroll
        for (int c = 0; c < DM; ++c) m += xs[tid][c];
        m *= (1.0f / DM);
        float var = 0.f;
#pragma unroll
        for (int c = 0; c < DM; ++c) { float d = xs[tid][c] - m; var += d * d; }
        var *= (1.0f / DM);
        float r = rsqrtf(var + 1e-5f);
        float xf[DM];
#pragma unroll
        for (int c = 0; c < DM; ++c)
            xf[c] = (xs[tid][c] - m) * r * lnf_w[c] + lnf_b[c];
        float* op = out + ((size_t)b * T + tid) * VOCAB;
#pragma unroll
        for (int v = 0; v < VOCAB; ++v) {
            float acc = 0.f;
#pragma unroll
            for (int j = 0; j < DM; ++j) acc += head_w[v * DM + j] * xf[j];
            op[v] = acc;
        }
    }
}

extern "C" void kernel_launch(void* const* d_in, const int* in_sizes, int n_in,
                              void* d_out, int out_size, void* d_ws, size_t ws_size,
                              hipStream_t stream) {
    const int*   idx     = (const int*)  d_in[0];
    const float* tok_emb = (const float*)d_in[1];
    const float* pos_enc = (const float*)d_in[2];
    const float* ln_w    = (const float*)d_in[3];
    const float* ln_b    = (const float*)d_in[4];
    const float* q1w     = (const float*)d_in[5];
    const float* k1w     = (const float*)d_in[6];
    const float* v1w     = (const float*)d_in[7];
    const float* q2w     = (const float*)d_in[8];
    const float* k2w     = (const float*)d_in[9];
    const float* v2w     = (const float*)d_in[10];
    const float* out_w   = (const float*)d_in[11];
    const float* lnf_w   = (const float*)d_in[12];
    const float* lnf_b   = (const float*)d_in[13];
    const float* head_w  = (const float*)d_in[14];
    float* out = (float*)d_out;

    const int batch = in_sizes[0] / T;   // 16384
    addtrans_kernel<<<batch, 64, 0, stream>>>(idx, tok_emb, pos_enc, ln_w, ln_b,
                                              q1w, k1w, v1w, q2w, k2w, v2w,
                                              out_w, lnf_w, lnf_b, head_w, out);
    (void)d_ws; (void)ws_size; (void)n_in; (void)out_size;
}